// RotationPositionLoss_45148696216032
// MI455X (gfx1250) — compile-verified
//
#include <hip/hip_runtime.h>
#include <math.h>

// ---------------------------------------------------------------------------
// RotationPositionLoss for MI455X (gfx1250, wave32)
// Streaming reduction: ~402 MB read once -> 1 scalar. HBM-bound (~17 us at
// 23.3 TB/s). Strategy: B128 non-temporal loads (read-once data, don't thrash
// the 192MB L2), hardware sqrt/rcp transcendentals, and a wave32 reduction
// done with a single V_WMMA_F32_16X16X4_F32 (A = [acc1|acc2] striped per the
// ISA 16x4 layout, B = 0/1 selector) that sums BOTH accumulators across all
// 32 lanes in one matrix op. Deterministic two-pass reduction (no atomics).
// ---------------------------------------------------------------------------

typedef float f32x4 __attribute__((ext_vector_type(4)));
typedef float v2f   __attribute__((ext_vector_type(2)));
typedef float v8f   __attribute__((ext_vector_type(8)));

#define NBLOCKS  4096
#define NTHREADS 256

__device__ __forceinline__ float clipf(float x, float lo, float hi) {
    return fminf(fmaxf(x, lo), hi);
}
__device__ __forceinline__ float sq(float x) { return x * x; }

struct V3 { float x, y, z; };

__device__ __forceinline__ float dot3(V3 a, V3 b) {
    return fmaf(a.x, b.x, fmaf(a.y, b.y, a.z * b.z));
}
__device__ __forceinline__ V3 cross3(V3 a, V3 b) {
    V3 c;
    c.x = fmaf(a.y, b.z, -a.z * b.y);
    c.y = fmaf(a.z, b.x, -a.x * b.z);
    c.z = fmaf(a.x, b.y, -a.y * b.x);
    return c;
}

// Gram-Schmidt on the columns of the row-major 3x3 matrix rc[9].
// Column c = (rc[c], rc[c+3], rc[c+6]). Matches the reference's
// u/(||u|| + 1e-8) normalization and det-based sign flip of column 3.
__device__ __forceinline__ void gram_schmidt(const float rc[9],
                                             V3& e1, V3& e2, V3& e3) {
    V3 u1 = { rc[0], rc[3], rc[6] };
    float inv1 = __builtin_amdgcn_rcpf(__builtin_amdgcn_sqrtf(dot3(u1, u1)) + 1e-8f);
    e1 = { u1.x * inv1, u1.y * inv1, u1.z * inv1 };

    V3 a2 = { rc[1], rc[4], rc[7] };
    float d12 = dot3(e1, a2);
    V3 u2 = { fmaf(-d12, e1.x, a2.x), fmaf(-d12, e1.y, a2.y), fmaf(-d12, e1.z, a2.z) };
    float inv2 = __builtin_amdgcn_rcpf(__builtin_amdgcn_sqrtf(dot3(u2, u2)) + 1e-8f);
    e2 = { u2.x * inv2, u2.y * inv2, u2.z * inv2 };

    V3 a3 = { rc[2], rc[5], rc[8] };
    float d13 = dot3(e1, a3);
    float d23 = dot3(e2, a3);
    V3 u3 = { fmaf(-d23, e2.x, fmaf(-d13, e1.x, a3.x)),
              fmaf(-d23, e2.y, fmaf(-d13, e1.y, a3.y)),
              fmaf(-d23, e2.z, fmaf(-d13, e1.z, a3.z)) };
    float inv3 = __builtin_amdgcn_rcpf(__builtin_amdgcn_sqrtf(dot3(u3, u3)) + 1e-8f);
    e3 = { u3.x * inv3, u3.y * inv3, u3.z * inv3 };

    // det([e1 e2 e3]) = (e1 x e2) . e3 ; flip third column if negative
    float det = dot3(cross3(e1, e2), e3);
    float sgn = det < 0.0f ? -1.0f : 1.0f;
    e3.x *= sgn; e3.y *= sgn; e3.z *= sgn;
}

__global__ __launch_bounds__(NTHREADS)
void rotpos_loss_stage1(const float* __restrict__ pred,
                        const float* __restrict__ tgt,
                        float* __restrict__ part1,  // weighted-loss partials
                        float* __restrict__ part2,  // raw-MSE partials (NaN fallback)
                        int B) {
    const int stride = (int)(gridDim.x * blockDim.x);
    float acc1 = 0.0f;   // sum of per-sample weighted loss contributions
    float acc2 = 0.0f;   // sum of per-sample 12-element squared diffs

    for (int s = (int)(blockIdx.x * blockDim.x + threadIdx.x); s < B; s += stride) {
        const f32x4* p4 = reinterpret_cast<const f32x4*>(pred) + 3 * (size_t)s;
        const f32x4* t4 = reinterpret_cast<const f32x4*>(tgt)  + 3 * (size_t)s;

        // Prefetch next grid-stride tile (gfx1250 global_prefetch_b8).
        if (s + stride < B) {
            __builtin_prefetch(p4 + 3 * (size_t)stride, 0, 1);
            __builtin_prefetch(t4 + 3 * (size_t)stride, 0, 1);
        }

        // Read-once streaming data: non-temporal B128 loads.
        f32x4 pa = __builtin_nontemporal_load(p4 + 0);
        f32x4 pb = __builtin_nontemporal_load(p4 + 1);
        f32x4 pc = __builtin_nontemporal_load(p4 + 2);
        f32x4 ta = __builtin_nontemporal_load(t4 + 0);
        f32x4 tb = __builtin_nontemporal_load(t4 + 1);
        f32x4 tc = __builtin_nontemporal_load(t4 + 2);

        const float pr[9] = { pa.x, pa.y, pa.z, pa.w, pb.x, pb.y, pb.z, pb.w, pc.x };
        const float tr[9] = { ta.x, ta.y, ta.z, ta.w, tb.x, tb.y, tb.z, tb.w, tc.x };

        float prc[9], trc[9];
#pragma unroll
        for (int i = 0; i < 9; ++i) {
            prc[i] = clipf(pr[i], -10.0f, 10.0f);
            trc[i] = clipf(tr[i], -10.0f, 10.0f);
        }

        V3 pe1, pe2, pe3, te1, te2, te3;
        gram_schmidt(prc, pe1, pe2, pe3);
        gram_schmidt(trc, te1, te2, te3);

        // chordal = ||Qp - Qt||_F^2 ; trace = <Qp, Qt>_F
        float chord =
            sq(pe1.x - te1.x) + sq(pe1.y - te1.y) + sq(pe1.z - te1.z) +
            sq(pe2.x - te2.x) + sq(pe2.y - te2.y) + sq(pe2.z - te2.z) +
            sq(pe3.x - te3.x) + sq(pe3.y - te3.y) + sq(pe3.z - te3.z);
        float trace = dot3(pe1, te1) + dot3(pe2, te2) + dot3(pe3, te3);

        trace = clipf(trace, -3.0f + 1e-6f, 3.0f - 1e-6f);
        float cosang = clipf((trace - 1.0f) * 0.5f, -1.0f + 1e-7f, 1.0f - 1e-7f);
        float ang = acosf(cosang) * 57.29577951308232f;
        ang = (ang != ang) ? 180.0f : ang;   // NaN -> 180

        // orthogonality on the clipped (pre-GS) pred 9d: ||M^T M - I||_F^2
        V3 c0 = { prc[0], prc[3], prc[6] };
        V3 c1 = { prc[1], prc[4], prc[7] };
        V3 c2 = { prc[2], prc[5], prc[8] };
        float d00 = dot3(c0, c0), d11 = dot3(c1, c1), d22 = dot3(c2, c2);
        float d01 = dot3(c0, c1), d02 = dot3(c0, c2), d12 = dot3(c1, c2);
        float ortho = sq(d00 - 1.0f) + sq(d11 - 1.0f) + sq(d22 - 1.0f)
                    + 2.0f * (sq(d01) + sq(d02) + sq(d12));

        // position MSE (pred position clipped, target raw)
        float pp0 = clipf(pc.y, -1000.0f, 1000.0f);
        float pp1 = clipf(pc.z, -1000.0f, 1000.0f);
        float pp2 = clipf(pc.w, -1000.0f, 1000.0f);
        float possq = sq(pp0 - tc.y) + sq(pp1 - tc.z) + sq(pp2 - tc.w);

        // L2 on raw pred 12d
        float l2 = sq(pa.x) + sq(pa.y) + sq(pa.z) + sq(pa.w)
                 + sq(pb.x) + sq(pb.y) + sq(pb.z) + sq(pb.w)
                 + sq(pc.x) + sq(pc.y) + sq(pc.z) + sq(pc.w);

        // raw 12d MSE (NaN fallback path)
        float mse = sq(pa.x - ta.x) + sq(pa.y - ta.y) + sq(pa.z - ta.z) + sq(pa.w - ta.w)
                  + sq(pb.x - tb.x) + sq(pb.y - tb.y) + sq(pb.z - tb.z) + sq(pb.w - tb.w)
                  + sq(pc.x - tc.x) + sq(pc.y - tc.y) + sq(pc.z - tc.z) + sq(pc.w - tc.w);

        // total_loss contribution: means fold into per-sample weights
        acc1 += chord + 0.1f * ang + 0.01f * ortho
              + (1.0f / 3.0f) * possq + (1e-4f / 12.0f) * l2;
        acc2 += mse;
    }

    // ------------------------------------------------------------------
    // Wave32 reduction of BOTH accumulators with one V_WMMA_F32_16X16X4_F32.
    // EXEC is all-ones here (no divergence after the loop).
    //
    // A (16x4, 2 VGPRs/lane, ISA layout):  A[m][0]=acc1[m], A[m][1]=acc2[m],
    //                                      A[m][2]=acc1[m+16], A[m][3]=acc2[m+16]
    // B (4x16, N = lane&15):               B[0][n]=B[2][n]= (n<8),
    //                                      B[1][n]=B[3][n]= (n>=8)
    // => D[m][n<8]  = acc1[m] + acc1[m+16]
    //    D[m][n>=8] = acc2[m] + acc2[m+16]
    // Summing the 8 D VGPRs per lane and folding lane halves gives the
    // full acc1 sum at lane 0 and the full acc2 sum at lane 8.
    // ------------------------------------------------------------------
    const int lane = threadIdx.x & 31;
    const int wid  = threadIdx.x >> 5;

    v2f A; A.x = acc1; A.y = acc2;
    float sel = ((lane & 15) < 8) ? 1.0f : 0.0f;
    v2f Bm; Bm.x = sel; Bm.y = 1.0f - sel;
    v8f C = {};
    v8f D = __builtin_amdgcn_wmma_f32_16x16x4_f32(
        /*neg_a=*/false, A, /*neg_b=*/false, Bm,
        /*c_mod=*/(short)0, C, /*reuse_a=*/false, /*reuse_b=*/false);

    float y = ((D[0] + D[1]) + (D[2] + D[3])) + ((D[4] + D[5]) + (D[6] + D[7]));
    y += __shfl_down(y, 16, 32);          // lanes 0-15: halves combined
    float w2 = __shfl(y, 8, 32);          // lane 8 holds the acc2 total

    __shared__ float s1[NTHREADS / 32];
    __shared__ float s2[NTHREADS / 32];
    if (lane == 0) { s1[wid] = y; s2[wid] = w2; }
    __syncthreads();
    if (wid == 0) {
        float a = (lane < NTHREADS / 32) ? s1[lane] : 0.0f;
        float b = (lane < NTHREADS / 32) ? s2[lane] : 0.0f;
#pragma unroll
        for (int off = 4; off > 0; off >>= 1) {
            a += __shfl_down(a, off, 32);
            b += __shfl_down(b, off, 32);
        }
        if (lane == 0) { part1[blockIdx.x] = a; part2[blockIdx.x] = b; }
    }
}

__global__ __launch_bounds__(256)
void rotpos_loss_stage2(const float* __restrict__ part1,
                        const float* __restrict__ part2,
                        int nblk, int B, float* __restrict__ out) {
    __shared__ double sh1[256];
    __shared__ double sh2[256];
    double a1 = 0.0, a2 = 0.0;
    for (int i = (int)threadIdx.x; i < nblk; i += 256) {
        a1 += (double)part1[i];
        a2 += (double)part2[i];
    }
    sh1[threadIdx.x] = a1;
    sh2[threadIdx.x] = a2;
    __syncthreads();
    for (int step = 128; step > 0; step >>= 1) {
        if ((int)threadIdx.x < step) {
            sh1[threadIdx.x] += sh1[threadIdx.x + step];
            sh2[threadIdx.x] += sh2[threadIdx.x + step];
        }
        __syncthreads();
    }
    if (threadIdx.x == 0) {
        double invB = 1.0 / (double)B;
        double total = sh1[0] * invB;
        double msefb = sh2[0] * invB / 12.0;
        out[0] = (total != total) ? (float)msefb : (float)total;
    }
}

extern "C" void kernel_launch(void* const* d_in, const int* in_sizes, int n_in,
                              void* d_out, int out_size, void* d_ws, size_t ws_size,
                              hipStream_t stream) {
    (void)n_in; (void)out_size; (void)ws_size;
    const float* pred = (const float*)d_in[0];
    const float* tgt  = (const float*)d_in[1];
    const int B = in_sizes[0] / 12;

    float* part1 = (float*)d_ws;          // NBLOCKS floats
    float* part2 = part1 + NBLOCKS;       // NBLOCKS floats (32 KB total)
    float* out   = (float*)d_out;

    rotpos_loss_stage1<<<NBLOCKS, NTHREADS, 0, stream>>>(pred, tgt, part1, part2, B);
    rotpos_loss_stage2<<<1, 256, 0, stream>>>(part1, part2, NBLOCKS, B, out);
}